// CrossAttentionProtoMatching_79731772883465
// MI455X (gfx1250) — compile-verified
//
#include <hip/hip_runtime.h>

typedef float v2f __attribute__((ext_vector_type(2)));
typedef float v8f __attribute__((ext_vector_type(8)));

constexpr int kB  = 4;
constexpr int kC  = 256;
constexpr int kHW = 4096;   // 64*64

// ---- async global->LDS helpers (CDNA5, ASYNCcnt-tracked) -------------------
__device__ __forceinline__ void async_load_b128(unsigned lds_byte_off, const void* gaddr) {
    asm volatile("global_load_async_to_lds_b128 %0, %1, off"
                 :: "v"(lds_byte_off), "v"(gaddr) : "memory");
}
__device__ __forceinline__ void wait_asynccnt_8() {
#if __has_builtin(__builtin_amdgcn_s_wait_asynccnt)
    __builtin_amdgcn_s_wait_asynccnt(8);
#else
    asm volatile("s_wait_asynccnt 0x8" ::: "memory");
#endif
}
__device__ __forceinline__ void wait_asynccnt_0() {
#if __has_builtin(__builtin_amdgcn_s_wait_asynccnt)
    __builtin_amdgcn_s_wait_asynccnt(0);
#else
    asm volatile("s_wait_asynccnt 0x0" ::: "memory");
#endif
}

// ---------------------------------------------------------------------------
// Kernel W: fold Wp/Wfc through Wv:  wpv[c] = sum_d Wp[d]*Wv[d,c], etc.
// ---------------------------------------------------------------------------
__global__ void k_weights(const float* __restrict__ Wv, const float* __restrict__ bv,
                          const float* __restrict__ Wp, const float* __restrict__ Wfc,
                          float* __restrict__ wpv, float* __restrict__ wfcv,
                          float* __restrict__ cst) {
    int c = threadIdx.x;                     // 256 threads
    float sp = 0.f, sf = 0.f;
    for (int d = 0; d < kC; ++d) {
        float wv = Wv[d * kC + c];
        sp += Wp[d]  * wv;
        sf += Wfc[d] * wv;
    }
    wpv[c]  = sp;
    wfcv[c] = sf;
    if (c == 0) {
        float a = 0.f, b2 = 0.f;
        for (int d = 0; d < kC; ++d) { a += Wp[d] * bv[d]; b2 += Wfc[d] * bv[d]; }
        cst[0] = a; cst[1] = b2;
    }
}

// ---------------------------------------------------------------------------
// Kernel P: Q/K projections with fp32 WMMA (M=16 positions, N=16 channels).
// ---------------------------------------------------------------------------
__global__ void __launch_bounds__(128)
k_proj(const float* __restrict__ qry, const float* __restrict__ supx,
       const float* __restrict__ Wq, const float* __restrict__ bq,
       const float* __restrict__ Wk, const float* __restrict__ bk,
       float* __restrict__ Qt, float* __restrict__ Kt) {
    const int wave = threadIdx.x >> 5;
    const int lane = threadIdx.x & 31;
    const int b     = blockIdx.y;
    const int which = blockIdx.z;                 // 0 -> Q, 1 -> K
    const int tile  = blockIdx.x * 4 + wave;
    const int dt = tile & 15;
    const int it = tile >> 4;
    const int i0 = it * 16, d0 = dt * 16;

    const float* X    = (which ? supx : qry) + (size_t)b * kC * kHW;
    const float* Wm   =  which ? Wk : Wq;
    const float* bias =  which ? bk : bq;
    float*       Out  = (which ? Kt : Qt) + (size_t)b * kHW * kC;

    const int half = lane >> 4;        // 0: K=0,1   1: K=2,3
    const int l    = lane & 15;
    const int mrow = i0 + l;           // A row (position)
    const int ncol = d0 + l;           // B/D column (out-channel)

    v8f acc = {};
    #pragma unroll
    for (int kc = 0; kc < kC; kc += 4) {
        const int ka = kc + half * 2;
        v2f a;
        a.x = X[(size_t)(ka + 0) * kHW + mrow];
        a.y = X[(size_t)(ka + 1) * kHW + mrow];
        v2f bw = *(const v2f*)&Wm[(size_t)ncol * kC + ka];
        acc = __builtin_amdgcn_wmma_f32_16x16x4_f32(false, a, false, bw,
                                                    (short)0, acc, false, false);
    }
    const float bb = bias[ncol];
    #pragma unroll
    for (int r = 0; r < 8; ++r) {
        const int row = i0 + r + half * 8;
        Out[(size_t)row * kC + ncol] = acc[r] + bb;
    }
}

// ---------------------------------------------------------------------------
// Kernel S: per-pixel pass: cosine sim + per-key scalars (mask, wpv.x, wfcv.x)
// ---------------------------------------------------------------------------
__global__ void k_pixel(const float* __restrict__ qry, const float* __restrict__ supx,
                        const float* __restrict__ supy,
                        const float* __restrict__ wpv, const float* __restrict__ wfcv,
                        const float* __restrict__ cst,
                        float4* __restrict__ keyscal, float* __restrict__ sim_out) {
    const int idx = blockIdx.x * blockDim.x + threadIdx.x;   // 0 .. B*HW-1
    if (idx >= kB * kHW) return;
    const int b = idx / kHW, i = idx % kHW;
    const float* xq = qry  + (size_t)b * kC * kHW + i;
    const float* xs = supx + (size_t)b * kC * kHW + i;
    float dot = 0.f, nq = 0.f, ns = 0.f, vp = 0.f, vf = 0.f;
    for (int c = 0; c < kC; ++c) {
        float a = xq[(size_t)c * kHW];
        float s = xs[(size_t)c * kHW];
        dot += a * s; nq += a * a; ns += s * s;
        vp += wpv[c] * s; vf += wfcv[c] * s;
    }
    keyscal[idx] = make_float4(supy[idx], vp + cst[0], vf + cst[1], 0.f);
    sim_out[idx] = dot / fmaxf(sqrtf(nq) * sqrtf(ns), 1e-8f);
}

// ---------------------------------------------------------------------------
// Kernel A: fused masked-softmax attention.  Each 128-thread block stages the
// shared 16-key tile (16KB, contiguous in the (HW,C) layout) into LDS via
// async global->LDS DMA, double-buffered so tile jt+1 streams in while the
// 4 waves run their 64 WMMAs on tile jt.  Score tile is M=keys x N=queries,
// so online softmax state is purely per-lane; one shfl_xor(16) merge at end.
// ---------------------------------------------------------------------------
__global__ void __launch_bounds__(128)
k_attn(const float* __restrict__ Qt, const float* __restrict__ Kt,
       const float4* __restrict__ keyscal, const float* __restrict__ supy,
       const float* __restrict__ bp, const float* __restrict__ bfc,
       float* __restrict__ out_raw, float* __restrict__ out_proto) {
    extern __shared__ char smem[];                 // 2 x 16KB key-tile buffers
    float* Ks = (float*)smem;

    const int wave = threadIdx.x >> 5;
    const int lane = threadIdx.x & 31;
    const int b  = blockIdx.y;
    const int i0 = blockIdx.x * 64 + wave * 16;
    const int half = lane >> 4;
    const int l    = lane & 15;

    const float*  Qb = Qt + (size_t)b * kHW * kC;
    const float*  Kb = Kt + (size_t)b * kHW * kC;
    const float4* ks = keyscal + (size_t)b * kHW;

    // B fragments: this wave's 16 queries x 256 channels (invariant over keys)
    v2f bf[64];
    const int qrow = i0 + l;
    #pragma unroll
    for (int kc = 0; kc < 64; ++kc)
        bf[kc] = *(const v2f*)&Qb[(size_t)qrow * kC + kc * 4 + half * 2];

    // issue one tile's async copy: 16 rows x 256 ch x 4B = 16KB, contiguous
    auto issue_tile = [&](int jt, int buf) {
        const char* gbase = (const char*)(Kb + (size_t)jt * 16 * kC) + threadIdx.x * 16;
        unsigned    lbase = (unsigned)(buf * 16384 + threadIdx.x * 16);
        #pragma unroll
        for (int q = 0; q < 8; ++q)                 // 8 x b128 per thread
            async_load_b128(lbase + q * 2048u, gbase + q * 2048);
    };

    constexpr int NT = kHW / 16;                    // 256 key tiles
    issue_tile(0, 0);

    float m = -3.0e38f, Z = 0.f, A = 0.f, Bp = 0.f, Bf = 0.f;

    for (int jt = 0; jt < NT; ++jt) {
        const int buf = jt & 1;
        if (jt + 1 < NT) {
            issue_tile(jt + 1, buf ^ 1);
            wait_asynccnt_8();                      // tile jt landed
        } else {
            wait_asynccnt_0();
        }
        __syncthreads();                            // tile jt visible to all waves

        const float* tile = Ks + buf * 4096;        // [16 keys][256 ch]
        v8f d = {};
        #pragma unroll
        for (int kc = 0; kc < 64; ++kc) {
            v2f a = *(const v2f*)&tile[l * kC + kc * 4 + half * 2];
            d = __builtin_amdgcn_wmma_f32_16x16x4_f32(false, a, false, bf[kc],
                                                      (short)0, d, false, false);
        }

        // online masked-softmax update (per lane: 8 keys for its query)
        float tm = d[0];
        #pragma unroll
        for (int r = 1; r < 8; ++r) tm = fmaxf(tm, d[r]);
        const float nm = fmaxf(m, tm);
        const float sc = __expf(m - nm);
        Z *= sc; A *= sc; Bp *= sc; Bf *= sc; m = nm;
        const int jb = jt * 16 + half * 8;
        #pragma unroll
        for (int r = 0; r < 8; ++r) {
            const float e = __expf(d[r] - nm);
            Z += e;
            const float4 s4 = ks[jb + r];
            const float em = e * s4.x;
            A += em; Bp += em * s4.y; Bf += em * s4.z;
        }
        __syncthreads();                            // reads done before buf reuse
    }

    // merge the two key-halves (lane L and L^16 hold the same query)
    {
        const float m2 = __shfl_xor(m, 16);
        const float nm = fmaxf(m, m2);
        const float s1 = __expf(m - nm), s2 = __expf(m2 - nm);
        Z  = Z  * s1 + __shfl_xor(Z, 16)  * s2;
        A  = A  * s1 + __shfl_xor(A, 16)  * s2;
        Bp = Bp * s1 + __shfl_xor(Bp, 16) * s2;
        Bf = Bf * s1 + __shfl_xor(Bf, 16) * s2;
    }

    if (half == 0) {
        const int i = i0 + l;
        const float den = fmaxf(A, 1e-12f * Z);     // softmax -> mask -> L1 norm
        const float inv = 1.f / den;
        const size_t o = (size_t)b * kHW + i;
        out_raw[o]   = Bp * inv + bp[0];
        out_proto[o] = supy[o] * (Bf * inv) + bfc[0];
    }
}

// ---------------------------------------------------------------------------
extern "C" void kernel_launch(void* const* d_in, const int* in_sizes, int n_in,
                              void* d_out, int out_size, void* d_ws, size_t ws_size,
                              hipStream_t stream) {
    const float* qry  = (const float*)d_in[0];
    const float* supx = (const float*)d_in[1];
    const float* supy = (const float*)d_in[2];
    const float* Wq   = (const float*)d_in[3];
    const float* bq   = (const float*)d_in[4];
    const float* Wk   = (const float*)d_in[5];
    const float* bk   = (const float*)d_in[6];
    const float* Wv   = (const float*)d_in[7];
    const float* bv   = (const float*)d_in[8];
    const float* Wfc  = (const float*)d_in[9];
    const float* bfc  = (const float*)d_in[10];
    const float* Wp   = (const float*)d_in[11];
    const float* bp   = (const float*)d_in[12];

    float* out_raw   = (float*)d_out;                 // (B,1,H,W)
    float* out_proto = out_raw + (size_t)kB * kHW;    // (B,1,H,W)
    float* out_sim   = out_raw + 2 * (size_t)kB * kHW;// (B,H,W)

    float*  ws      = (float*)d_ws;
    float*  Qt      = ws;                                       // B*HW*C
    float*  Kt      = ws + (size_t)kB * kHW * kC;               // B*HW*C
    float*  ksf     = ws + 2 * (size_t)kB * kHW * kC;           // B*HW*4
    float4* keyscal = (float4*)ksf;
    float*  wpv     = ksf + (size_t)kB * kHW * 4;               // C
    float*  wfcv    = wpv + kC;                                 // C
    float*  cst     = wfcv + kC;                                // 2

    k_weights<<<1, kC, 0, stream>>>(Wv, bv, Wp, Wfc, wpv, wfcv, cst);

    k_proj<<<dim3((kHW / 16) * (kC / 16) / 4, kB, 2), 128, 0, stream>>>(
        qry, supx, Wq, bq, Wk, bk, Qt, Kt);

    k_pixel<<<dim3((kB * kHW) / 256), 256, 0, stream>>>(
        qry, supx, supy, wpv, wfcv, cst, keyscal, out_sim);

    k_attn<<<dim3(kHW / 64, kB), 128, 32768, stream>>>(
        Qt, Kt, keyscal, supy, bp, bfc, out_raw, out_proto);
}